// SAGScorer_14044543057996
// MI455X (gfx1250) — compile-verified
//
#include <hip/hip_runtime.h>
#include <hip/hip_bf16.h>
#include <cstdint>

#define DFEAT 64

typedef float v2f __attribute__((ext_vector_type(2)));
typedef float v8f __attribute__((ext_vector_type(8)));

// ---------------------------------------------------------------------------
// 1) init: deg = 1 (self loop), h1 = 0, h2 = 0   (re-run every call: graph-safe)
// ---------------------------------------------------------------------------
__global__ void sag_init_kernel(float* __restrict__ deg,
                                float4* __restrict__ h1,
                                float4* __restrict__ h2, int n) {
  const int i = blockIdx.x * blockDim.x + threadIdx.x;
  if (i < n) deg[i] = 1.0f;                    // self-loop contributes 1
  const int total4 = n * (DFEAT / 4);
  if (i < total4) {
    const float4 z = make_float4(0.f, 0.f, 0.f, 0.f);
    h1[i] = z;
    h2[i] = z;
  }
}

// ---------------------------------------------------------------------------
// 2) degree accumulate over dst  (L2-resident f32 atomics)
// ---------------------------------------------------------------------------
__global__ void sag_deg_kernel(const int* __restrict__ dst,
                               float* __restrict__ deg, int e) {
  const int i = blockIdx.x * blockDim.x + threadIdx.x;
  if (i < e) unsafeAtomicAdd(&deg[dst[i]], 1.0f);
}

// ---------------------------------------------------------------------------
// 3) dinv = rsqrt(deg)  (deg >= 1 always, matches reference's deg>0 guard)
// ---------------------------------------------------------------------------
__global__ void sag_dinv_kernel(float* __restrict__ deg, int n) {
  const int i = blockIdx.x * blockDim.x + threadIdx.x;
  if (i < n) deg[i] = rsqrtf(deg[i]);
}

// ---------------------------------------------------------------------------
// 4/5) one propagation hop: hout[dst] += dinv[src]*dinv[dst] * hin[src]
//      16 lanes per (edge|self-loop); each lane: float4 gather + 4 f32 atomics
// ---------------------------------------------------------------------------
__global__ void sag_hop_kernel(const float* __restrict__ hin,
                               float* __restrict__ hout,
                               const int* __restrict__ src,
                               const int* __restrict__ dst,
                               const float* __restrict__ dinv,
                               int e, int n) {
  const int tid  = blockIdx.x * blockDim.x + threadIdx.x;
  const int work = tid >> 4;   // edge id (or self-loop id past e)
  const int lane = tid & 15;   // 16 lanes * float4 = 64 channels
  if (work >= e + n) return;

  int s, t;
  if (work < e) { s = src[work]; t = dst[work]; }
  else          { s = t = work - e; }           // self loop

  const float coef = dinv[s] * dinv[t];
  const float4 v =
      *reinterpret_cast<const float4*>(hin + (size_t)s * DFEAT + lane * 4);
  float* o = hout + (size_t)t * DFEAT + lane * 4;
  unsafeAtomicAdd(o + 0, coef * v.x);
  unsafeAtomicAdd(o + 1, coef * v.y);
  unsafeAtomicAdd(o + 2, coef * v.z);
  unsafeAtomicAdd(o + 3, coef * v.w);
}

// ---------------------------------------------------------------------------
// 6) score = sigmoid(h2 @ W + b) via V_WMMA_F32_16X16X4_F32 (16 nodes / wave),
//    then out_x = x * score (in place over h2's region) and score tail.
//    A tile: lanes 0-15 -> K={k0,k0+1}; lanes 16-31 -> K={k0+2,k0+3} (ISA layout).
//    B tile: W chunk broadcast to all 16 N-columns -> every D column = h.W dot.
// ---------------------------------------------------------------------------
__global__ void sag_score_kernel(const float* __restrict__ x,
                                 const float* __restrict__ h2,
                                 const float* __restrict__ W,
                                 const float* __restrict__ bptr,
                                 float* __restrict__ outx,
                                 float* __restrict__ outs, int n) {
  const int lane = threadIdx.x & 31;
  const int wave = threadIdx.x >> 5;
  const int nb   = (blockIdx.x * (blockDim.x >> 5) + wave) * 16;
  if (nb >= n) return;                       // wave-uniform exit: EXEC stays all-1s

  const int half = lane >> 4;                // which K pair this lane supplies
  const int m    = lane & 15;                // row within the 16-node tile
  int rowNode = nb + m;
  if (rowNode > n - 1) rowNode = n - 1;      // clamp (keeps EXEC full for WMMA)
  const float* hrow = h2 + (size_t)rowNode * DFEAT;

  v8f c = {};
#pragma unroll
  for (int k0 = 0; k0 < DFEAT; k0 += 4) {
    const int kb = k0 + half * 2;
    v2f a;  a.x  = hrow[kb]; a.y  = hrow[kb + 1];
    v2f bm; bm.x = W[kb];    bm.y = W[kb + 1];   // N-invariant broadcast of W
    c = __builtin_amdgcn_wmma_f32_16x16x4_f32(
        /*neg_a=*/false, a, /*neg_b=*/false, bm,
        /*c_mod=*/(short)0, c, /*reuse_a=*/false, /*reuse_b=*/false);
  }

  // C/D layout: lanes 0-15 VGPR r -> M=r ; lanes 16-31 VGPR r -> M=8+r.
  // All 16 columns identical, so every lane in a half holds its 8 node dots.
  const float bias = bptr[0];
#pragma unroll
  for (int r = 0; r < 8; ++r) {
    const int node = nb + half * 8 + r;
    if (node < n) {
      const float sc = 1.0f / (1.0f + __expf(-(c[r] + bias)));
      if (m == 0) outs[node] = sc;                       // lanes 0 & 16 write tail
      // 16 lanes of this half cooperatively write the 64-wide row as float4s
      const float4 xv =
          reinterpret_cast<const float4*>(x + (size_t)node * DFEAT)[m];
      const float4 ov = make_float4(xv.x * sc, xv.y * sc, xv.z * sc, xv.w * sc);
      reinterpret_cast<float4*>(outx + (size_t)node * DFEAT)[m] = ov;
    }
  }
}

// ---------------------------------------------------------------------------
extern "C" void kernel_launch(void* const* d_in, const int* in_sizes, int n_in,
                              void* d_out, int out_size, void* d_ws, size_t ws_size,
                              hipStream_t stream) {
  const float* x  = (const float*)d_in[0];
  const int*   ei = (const int*)d_in[1];
  const float* W  = (const float*)d_in[2];
  const float* b  = (const float*)d_in[3];

  const int n = in_sizes[0] / DFEAT;   // 100000
  const int e = in_sizes[1] / 2;       // 1600000
  const int* src = ei;                 // edge_index[0]
  const int* dst = ei + e;             // edge_index[1]

  // Workspace: [deg/dinv : nPad floats][h1 : n*64 floats]  (~26 MB)
  float* deg = (float*)d_ws;
  const size_t nPad = ((size_t)n + 63) & ~(size_t)63;   // keep h1 256B-aligned
  float* h1 = deg + nPad;
  // h2 reuses the x*score region of d_out (read-before-overwrite in score kernel)
  float* h2   = (float*)d_out;
  float* outs = (float*)d_out + (size_t)n * DFEAT;

  const int B = 256;

  {  // 1) init accumulators (must run every call — graph replays)
    const int total = n * (DFEAT / 4);
    sag_init_kernel<<<(total + B - 1) / B, B, 0, stream>>>(
        deg, (float4*)h1, (float4*)h2, n);
  }
  // 2) degree
  sag_deg_kernel<<<(e + B - 1) / B, B, 0, stream>>>(dst, deg, e);
  // 3) dinv (in place)
  sag_dinv_kernel<<<(n + B - 1) / B, B, 0, stream>>>(deg, n);
  // 4) hop 1: x -> h1
  {
    const long long work = (long long)(e + n) * 16;
    sag_hop_kernel<<<(int)((work + B - 1) / B), B, 0, stream>>>(
        x, h1, src, dst, deg, e, n);
  }
  // 5) hop 2: h1 -> h2 (d_out region)
  {
    const long long work = (long long)(e + n) * 16;
    sag_hop_kernel<<<(int)((work + B - 1) / B), B, 0, stream>>>(
        h1, h2, src, dst, deg, e, n);
  }
  // 6) WMMA scoring + finalize (outx overlays h2 region, safe in-wave)
  {
    const int nodesPerBlock = (B / 32) * 16;  // 8 waves * 16 nodes = 128
    sag_score_kernel<<<(n + nodesPerBlock - 1) / nodesPerBlock, B, 0, stream>>>(
        x, h2, W, b, /*outx=*/h2, outs, n);
  }
}